// ParallelRYComplex_30494267801965
// MI455X (gfx1250) — compile-verified
//
#include <hip/hip_runtime.h>

// ParallelRY on 13 qubits: out = x @ kron(G0..G12), G = [[c,-s],[s,c]], c/s = cos/sin(theta/2).
// Factorized: 3x (4-qubit group as 16x16 matrix applied via chained V_WMMA_F32_16X16X4_F32)
// + 1 scalar butterfly for the MSB. Whole 8192-float row lives in LDS; one HBM pass.

typedef __attribute__((ext_vector_type(2))) float v2f;
typedef __attribute__((ext_vector_type(8))) float v8f;

#define DIM 8192
#define NQ  13

// One 16x16 f32 GEMM tile:  D[m][n] = sum_k A[m][k] * B16[k][n]
// A element (m,k) lives at A[m*mS + k*kS] in LDS; B16 is a row-major 16x16 LDS matrix.
// Chained as 4x V_WMMA_F32_16X16X4_F32 (K=4 each), f32 exact.
__device__ __forceinline__ v8f wmma16x16(const float* __restrict__ A, int mS, int kS,
                                         const float* __restrict__ B16, int lane)
{
    const int m  = lane & 15;          // A: M index striped across lanes 0-15 / 16-31
    const int n  = lane & 15;          // B: N index striped across lanes
    const int kh = (lane >> 4) << 1;   // lane-half selects K pair {0,1} vs {2,3}
    v8f acc = {};
    #pragma unroll
    for (int k4 = 0; k4 < 16; k4 += 4) {
        const int k0 = k4 + kh;
        v2f a, b;
        a.x = A[m * mS + (k0    ) * kS];
        a.y = A[m * mS + (k0 + 1) * kS];
        b.x = B16[(k0    ) * 16 + n];
        b.y = B16[(k0 + 1) * 16 + n];
        acc = __builtin_amdgcn_wmma_f32_16x16x4_f32(
            /*neg_a=*/false, a, /*neg_b=*/false, b,
            /*c_mod=*/(short)0, acc, /*reuse_a=*/false, /*reuse_b=*/false);
    }
    return acc;
}

// Scatter D (8 VGPRs, M = r + 8*(lane>=16), N = lane&15) back to LDS at D[m*mS + n*nS].
__device__ __forceinline__ void wmma_store(float* __restrict__ D, int mS, int nS,
                                           v8f acc, int lane)
{
    const int n   = lane & 15;
    const int mhi = (lane >> 4) << 3;  // 0 or 8
    #pragma unroll
    for (int r = 0; r < 8; ++r)
        D[(r + mhi) * mS + n * nS] = acc[r];
}

__global__ __launch_bounds__(256)
void ry13_apply_kernel(const float* __restrict__ params,
                       const float* __restrict__ x_real,
                       const float* __restrict__ x_imag,
                       float* __restrict__ out)
{
    __shared__ float row[DIM];       // 32 KB: one batch row
    __shared__ float M[3][256];      // three 16x16 group matrices (bits 0-3, 4-7, 8-11)
    __shared__ float cs_l[NQ], sn_l[NQ];

    const int tid  = threadIdx.x;
    const int lane = tid & 31;
    const int wave = tid >> 5;

    // --- 1. half-angle cos/sin ------------------------------------------------
    if (tid < NQ) {
        const float h = 0.5f * params[tid];
        cs_l[tid] = cosf(h);
        sn_l[tid] = sinf(h);
    }

    // --- 3a. start the global->LDS row load (overlaps cos/sin) ----------
    const float* __restrict__ src =
        (blockIdx.x < 16) ? (x_real + (size_t)blockIdx.x * DIM)
                          : (x_imag + (size_t)(blockIdx.x - 16) * DIM);
    {
        const float4* sv = (const float4*)src;
        float4* rv = (float4*)row;
        #pragma unroll
        for (int k = 0; k < DIM / 4 / 256; ++k)    // 8 x float4 per thread
            rv[tid + k * 256] = sv[tid + k * 256];
    }
    __syncthreads();   // cs/sn visible

    // --- 2. build the three 16x16 group matrices ------------------------------
    // Global bit p (0 = LSB) carries gate gates[12 - p] (gates[0] is the kron MSB).
    // G[ib][jb]: ib==jb -> c ; ib<jb -> -s ; ib>jb -> s.
    {
        const int i = tid >> 4, j = tid & 15;
        #pragma unroll
        for (int g = 0; g < 3; ++g) {
            float prod = 1.0f;
            #pragma unroll
            for (int b = 0; b < 4; ++b) {
                const int q  = 12 - (g * 4 + b);
                const float c = cs_l[q], s = sn_l[q];
                const int ib = (i >> b) & 1, jb = (j >> b) & 1;
                prod *= (ib == jb) ? c : ((ib == 0) ? -s : s);
            }
            M[g][i * 16 + j] = prod;
        }
    }
    __syncthreads();   // row + matrices ready

    // --- Pass A: bits 0-3 (contiguous groups of 16) ---------------------------
    // Row viewed as [512][16]; tile = 16 consecutive group-vectors -> 16x16 GEMM.
    #pragma unroll
    for (int t = 0; t < 4; ++t) {
        float* base = row + (wave * 4 + t) * 256;
        v8f acc = wmma16x16(base, /*mS=*/16, /*kS=*/1, M[0], lane);
        wmma_store(base, 16, 1, acc, lane);
    }
    __syncthreads();

    // --- Pass B: bits 4-7 (contraction stride 16) -----------------------------
    // Row viewed as [32][16][16]; per hi-block, free index = low 4 bits.
    #pragma unroll
    for (int t = 0; t < 4; ++t) {
        float* base = row + (wave * 4 + t) * 256;
        v8f acc = wmma16x16(base, /*mS=*/1, /*kS=*/16, M[1], lane);
        wmma_store(base, 1, 16, acc, lane);
    }
    __syncthreads();

    // --- Pass C: bits 8-11 (contraction stride 256) ---------------------------
    // Row viewed as [2][16][16][16]; tile = (top, low-group-of-16).
    #pragma unroll
    for (int t = 0; t < 4; ++t) {
        const int tt = wave * 4 + t;
        float* base = row + (tt >> 4) * 4096 + (tt & 15) * 16;
        v8f acc = wmma16x16(base, /*mS=*/1, /*kS=*/256, M[2], lane);
        wmma_store(base, 1, 256, acc, lane);
    }
    __syncthreads();

    // --- Pass D: bit 12 (MSB) butterfly, gate = gates[0] ----------------------
    {
        const float c = cs_l[0], s = sn_l[0];
        #pragma unroll
        for (int it = 0; it < 16; ++it) {
            const int k = tid + it * 256;          // k in [0, 4096)
            const float a = row[k], b = row[k + 4096];
            row[k]        =  c * a + s * b;
            row[k + 4096] = -s * a + c * b;
        }
    }
    __syncthreads();

    // --- store ---------------------------------------------------------------
    {
        float4* dv = (float4*)(out + (size_t)blockIdx.x * DIM);
        const float4* rv = (const float4*)row;
        #pragma unroll
        for (int k = 0; k < DIM / 4 / 256; ++k)
            dv[tid + k * 256] = rv[tid + k * 256];
    }
}

extern "C" void kernel_launch(void* const* d_in, const int* in_sizes, int n_in,
                              void* d_out, int out_size, void* d_ws, size_t ws_size,
                              hipStream_t stream)
{
    const float* params = (const float*)d_in[0];   // [13]
    const float* x_real = (const float*)d_in[1];   // [16, 8192]
    const float* x_imag = (const float*)d_in[2];   // [16, 8192]
    float* out = (float*)d_out;                    // [2, 16, 8192] concatenated

    // 32 blocks: one per (real|imag) batch row; 256 threads = 8 wave32.
    hipLaunchKernelGGL(ry13_apply_kernel, dim3(32), dim3(256), 0, stream,
                       params, x_real, x_imag, out);
}